// Attention_64682207478593
// MI455X (gfx1250) — compile-verified
//
#include <hip/hip_runtime.h>
#include <hip/hip_bf16.h>

// ---------------------------------------------------------------------------
// MI455X (gfx1250) attention layer.
// Bandwidth-bound (~0.7 GB streamed @ 23.3 TB/s => ~30us floor, ~3.7 GFLOP):
// fp32 V_WMMA_F32_16X16X4_F32 for every matmul; Tensor Data Mover (TDM)
// double-buffers ALL big streams (weights, K cache, V cache) into LDS.
// ---------------------------------------------------------------------------

typedef float v2f __attribute__((ext_vector_type(2)));
typedef float v8f __attribute__((ext_vector_type(8)));
typedef unsigned int u32x4 __attribute__((ext_vector_type(4)));
typedef int i32x4 __attribute__((ext_vector_type(4)));
typedef int i32x8 __attribute__((ext_vector_type(8)));

#define BATCH 8
#define SEQ   4
#define TCTX  2048
#define NHEAD 32
#define NKVH  8
#define HDIM  128
#define DIM   4096
#define MTOK  32            // BATCH*SEQ
#define KVDIM 1024          // NKVH*HDIM
#define NTOT  6144          // DIM + 2*KVDIM  (fused Q|K|V)

#define LPITCH 68           // 64-col LDS pitch: 68%64==4 -> conflict-free
#define KPITCH 132          // 128-col LDS pitch: 132%64==4 -> conflict-free
#define QPITCH 132

static __device__ __forceinline__ v8f wmma_f32(v2f a, v2f b, v8f c) {
    return __builtin_amdgcn_wmma_f32_16x16x4_f32(false, a, false, b, (short)0, c,
                                                 false, false);
}

// ---------------------------------------------------------------------------
// TDM: DMA a rows x cols fp32 tile (row stride `stride` floats) into LDS.
// pad_amount = 4 DWORDs every 2^(pad_iv+1) DWORDs -> pitch cols+4 floats
// (pad_iv=5 for 64 cols, 6 for 128 cols).  D# per CDNA5 ISA §8.3-8.6.
// TENSORcnt is wave-private -> one wave issues, then barrier publishes.
// ---------------------------------------------------------------------------
static __device__ __forceinline__ void tdm_load_2d(
    const float* gsrc, unsigned int lds_byte_off,
    unsigned int cols, unsigned int rows, unsigned long long stride,
    unsigned int pad_iv)
{
    unsigned long long ga = (unsigned long long)(uintptr_t)gsrc;
    u32x4 g0;
    g0[0] = 1u;                                        // count=1, user mode
    g0[1] = lds_byte_off;                              // LDS dest (bytes)
    g0[2] = (unsigned int)ga;                          // global_addr[31:0]
    g0[3] = (unsigned int)((ga >> 32) & 0x1FFFFFFu)    // global_addr[56:32]
          | (2u << 30);                                // type=2 ("image")

    const unsigned long long td0 = cols;               // tensor_dim0
    const unsigned long long td1 = rows;               // tensor_dim1
    unsigned long long q0 = ((2ull << 16)              // data_size = 4B
                          |  (1ull << 20)              // pad_enable
                          |  ((unsigned long long)pad_iv << 22)
                          |  (3ull << 25))             // pad_amount: 4 DW
                          | ((td0 & 0xFFFFull) << 48); // tensor_dim0[15:0]
    unsigned long long q1 = (td0 >> 16)                // tensor_dim0[31:16]
                          | (td1 << 16)                // tensor_dim1 @ [111:80]
                          | ((unsigned long long)cols << 48);  // tile_dim0
    unsigned long long q2 = (unsigned long long)rows   // tile_dim1 @ [143:128]
                          | ((stride & 0xFFFFFFFFull) << 32);  // stride0[31:0]
    unsigned long long q3 = (stride >> 32);            // stride0[47:32]

    i32x8 g1;
    g1[0] = (int)(unsigned int)q0; g1[1] = (int)(unsigned int)(q0 >> 32);
    g1[2] = (int)(unsigned int)q1; g1[3] = (int)(unsigned int)(q1 >> 32);
    g1[4] = (int)(unsigned int)q2; g1[5] = (int)(unsigned int)(q2 >> 32);
    g1[6] = (int)(unsigned int)q3; g1[7] = (int)(unsigned int)(q3 >> 32);

    i32x4 gz = {0, 0, 0, 0};
#if defined(__clang_major__) && (__clang_major__ >= 23)
    i32x8 gz8 = {0, 0, 0, 0, 0, 0, 0, 0};
    __builtin_amdgcn_tensor_load_to_lds(g0, g1, gz, gz, gz8, 0);
#else
    __builtin_amdgcn_tensor_load_to_lds(g0, g1, gz, gz, 0);
#endif
}

// ---------------------------------------------------------------------------
// Fused QKV projection: [32 x 4096] x {wq^T | wk^T | wv^T}.
// grid.x = 96, block = 256 (8 waves).  TDM double-buffers the weight tiles.
// ---------------------------------------------------------------------------
__global__ __launch_bounds__(256) void qkv_gemm(
    const float* __restrict__ x,
    const float* __restrict__ wq, const float* __restrict__ wk,
    const float* __restrict__ wv,
    float* __restrict__ qo, float* __restrict__ ko, float* __restrict__ vo)
{
    __shared__ float a_lds[MTOK * LPITCH];
    __shared__ float b_lds[2][64 * LPITCH];

    const int tid  = threadIdx.x;
    const int lane = tid & 31;
    const int wave = tid >> 5;
    const int mt   = wave >> 2, nt = wave & 3;
    const int n_base = blockIdx.x * 64;

    const float* wbase;
    if (n_base < DIM)            wbase = wq + (size_t)n_base * DIM;
    else if (n_base < DIM+KVDIM) wbase = wk + (size_t)(n_base - DIM) * DIM;
    else                         wbase = wv + (size_t)(n_base - DIM - KVDIM) * DIM;

    const unsigned int ldsb0 = (unsigned int)(uintptr_t)&b_lds[0][0];
    const unsigned int ldsb1 = (unsigned int)(uintptr_t)&b_lds[1][0];

    const int arow = tid >> 3;
    const int acol = (tid & 7) * 8;
    const int afrag = (mt*16 + (lane & 15)) * LPITCH + 2*(lane >> 4);
    const int bfrag = (nt*16 + (lane & 15)) * LPITCH + 2*(lane >> 4);

    if (wave == 0)
        tdm_load_2d(wbase, ldsb0, 64, 64, DIM, 5);

    v8f acc = {};
    int cur = 0;
    for (int k0 = 0; k0 < DIM; k0 += 64) {
        __syncthreads();
        {
            const float4* s = (const float4*)(x + (size_t)arow * DIM + k0 + acol);
            float4 p0 = s[0], p1 = s[1];
            *(float4*)(a_lds + arow*LPITCH + acol)     = p0;
            *(float4*)(a_lds + arow*LPITCH + acol + 4) = p1;
            if (k0 + 64 < DIM)
                __builtin_prefetch(x + (size_t)arow * DIM + k0 + 64 + acol, 0, 1);
        }
        if (wave == 0) {
            if (k0 + 64 < DIM) {
                tdm_load_2d(wbase + k0 + 64, cur ? ldsb0 : ldsb1, 64, 64, DIM, 5);
                __builtin_amdgcn_s_wait_tensorcnt(1);
            } else {
                __builtin_amdgcn_s_wait_tensorcnt(0);
            }
        }
        __syncthreads();
        const float* bb = b_lds[cur];
        #pragma unroll
        for (int kk = 0; kk < 64; kk += 4) {
            v2f a = *(const v2f*)(a_lds + afrag + kk);
            v2f b = *(const v2f*)(bb    + bfrag + kk);
            acc = wmma_f32(a, b, acc);
        }
        cur ^= 1;
    }

    const int n   = n_base + nt*16 + (lane & 15);
    const int mb  = mt*16 + 8*(lane >> 4);
    float* optr; int ncol, opitch;
    if (n < DIM)            { optr = qo; ncol = n;               opitch = DIM;   }
    else if (n < DIM+KVDIM) { optr = ko; ncol = n - DIM;         opitch = KVDIM; }
    else                    { optr = vo; ncol = n - DIM - KVDIM; opitch = KVDIM; }
    #pragma unroll
    for (int r = 0; r < 8; ++r)
        optr[(size_t)(mb + r) * opitch + ncol] = acc[r];
}

// ---------------------------------------------------------------------------
// RoPE on q (32 heads) and k (8 heads).
// ---------------------------------------------------------------------------
__global__ __launch_bounds__(256) void rope_kernel(
    float* __restrict__ q, float* __restrict__ k,
    const float* __restrict__ angles)
{
    const int idx = blockIdx.x * blockDim.x + threadIdx.x;
    const int NPAIR = MTOK * (NHEAD + NKVH) * (HDIM / 2);  // 81920
    if (idx >= NPAIR) return;
    const int j = idx & 63;
    const int r = idx >> 6;
    const int h = r % (NHEAD + NKVH);
    const int m = r / (NHEAD + NKVH);
    const float a = angles[m * 64 + j];
    const float c = cosf(a), s = sinf(a);
    float* base = (h < NHEAD) ? (q + (size_t)m * DIM   + h * HDIM)
                              : (k + (size_t)m * KVDIM + (h - NHEAD) * HDIM);
    const float xr = base[2*j], xi = base[2*j + 1];
    base[2*j]     = xr * c - xi * s;
    base[2*j + 1] = xr * s + xi * c;
}

// ---------------------------------------------------------------------------
// Attention core: one workgroup per (b,h). grid = 256, block = 256 (8 waves).
// Phase 1: TDM double-buffers 128x128 K tiles into LDS; wave 0 patches the
//          <=4 rows in [sp, sp+4) with fresh K; each wave WMMAs 16 t-rows.
// Phase 2: softmax (waves 0..3, shfl_xor reductions); first V tile DMAs
//          concurrently.
// Phase 3: out^T = V^T.P^T; TDM double-buffers 64x128 V tiles (same LDS).
// ---------------------------------------------------------------------------
__global__ __launch_bounds__(256) void attn_kernel(
    const float* __restrict__ q_ws, const float* __restrict__ k_ws,
    const float* __restrict__ v_ws,
    const float* __restrict__ cache_k, const float* __restrict__ cache_v,
    const float* __restrict__ mask, const int* __restrict__ start_pos,
    const int* __restrict__ layer_idx, float* __restrict__ attn_out)
{
    __shared__ float kv_lds[2][128 * KPITCH];   // K tiles; reused for V tiles
    __shared__ float q_lds[16 * QPITCH];
    __shared__ float sc_lds[TCTX * 4 + 16];

    const int bh = blockIdx.x;
    const int b  = bh >> 5, h = bh & 31, kvh = h >> 2;
    const int tid = threadIdx.x, lane = tid & 31, wave = tid >> 5;
    const int sp   = start_pos[b];
    const int lidx = layer_idx[0];

    const unsigned int kb0 = (unsigned int)(uintptr_t)&kv_lds[0][0];
    const unsigned int kb1 = (unsigned int)(uintptr_t)&kv_lds[1][0];
    // V double-buffer lives inside kv_lds[0] (64*KPITCH each half)
    float* const vbuf0 = &kv_lds[0][0];
    float* const vbuf1 = &kv_lds[0][64 * KPITCH];
    const unsigned int vb0 = (unsigned int)(uintptr_t)vbuf0;
    const unsigned int vb1 = (unsigned int)(uintptr_t)vbuf1;

    // q tile (4 x 128) -> LDS rows 0..3 (rows 4..15 feed discarded D rows)
    {
        const int s = tid >> 6, d = (tid & 63) * 2;
        if (s < SEQ) {
            float2 v = *(const float2*)(q_ws + (size_t)(b*SEQ + s) * DIM + h*HDIM + d);
            *(float2*)(q_lds + s * QPITCH + d) = v;
        }
    }

    const int l15  = lane & 15;
    const int dsel = 2 * (lane >> 4);

    const float* kc0 = cache_k + ((size_t)(b * TCTX) * 2 + lidx) * DIM + h * HDIM;
    const float* vc0 = cache_v + ((size_t)(b * TCTX) * 2 + lidx) * DIM + h * HDIM;
    const float* kf0 = k_ws + (size_t)(b * SEQ) * KVDIM + kvh * HDIM;
    const float* vf0 = v_ws + (size_t)(b * SEQ) * KVDIM + kvh * HDIM;

    // ---------------- Phase 1: scores ----------------
    if (wave == 0)
        tdm_load_2d(kc0, kb0, 128, 128, 2 * DIM, 6);

    int cur = 0;
    for (int t0 = 0; t0 < TCTX; t0 += 128) {
        __syncthreads();                          // prior tile readers done
        if (wave == 0) {
            if (t0 + 128 < TCTX) {
                tdm_load_2d(kc0 + (size_t)(t0 + 128) * 2 * DIM,
                            cur ? kb0 : kb1, 128, 128, 2 * DIM, 6);
                __builtin_amdgcn_s_wait_tensorcnt(1);
            } else {
                __builtin_amdgcn_s_wait_tensorcnt(0);
            }
            // patch rows in the scatter window with fresh (RoPE'd) K
            const int lo = (t0 > sp) ? t0 : sp;
            const int hi = ((t0 + 128) < (sp + SEQ)) ? (t0 + 128) : (sp + SEQ);
            for (int t = lo; t < hi; ++t) {
                float4 vv = *(const float4*)(kf0 + (size_t)(t - sp) * KVDIM + lane*4);
                *(float4*)(&kv_lds[cur][(t - t0) * KPITCH + lane*4]) = vv;
            }
        }
        __syncthreads();                          // publish tile
        const float* kt = &kv_lds[cur][(wave * 16) * KPITCH];
        const int tl = t0 + wave * 16 + l15;      // B-fragment column (N)
        v8f c = {};
        #pragma unroll
        for (int d = 0; d < HDIM; d += 4) {
            v2f a  = *(const v2f*)(q_lds + l15 * QPITCH + d + dsel);
            v2f bv = *(const v2f*)(kt    + l15 * KPITCH + d + dsel);
            c = wmma_f32(a, bv, c);
        }
        if (lane < 16) {                          // D rows 0..3 = the 4 queries
            float4 sv = { c[0], c[1], c[2], c[3] };
            *(float4*)(sc_lds + tl * 4) = sv;
        }
        cur ^= 1;
    }
    __syncthreads();

    // Prefetch first V tile while softmax runs (phase-1 LDS reads are done).
    if (wave == 0)
        tdm_load_2d(vc0, vb0, 128, 64, 2 * DIM, 6);

    // ---------------- Phase 2: softmax ----------------
    if (wave < 4) {
        const int  qr    = wave;
        const float scale = 0.08838834764831845f;   // 1/sqrt(128)
        float mx = -3.0e38f;
        for (int i = lane; i < TCTX; i += 32)
            mx = fmaxf(mx, sc_lds[i*4 + qr] * scale + mask[qr * TCTX + i]);
        #pragma unroll
        for (int o = 16; o; o >>= 1) mx = fmaxf(mx, __shfl_xor(mx, o, 32));
        float sum = 0.f;
        for (int i = lane; i < TCTX; i += 32) {
            float e = __expf(sc_lds[i*4 + qr] * scale + mask[qr * TCTX + i] - mx);
            sc_lds[i*4 + qr] = e;
            sum += e;
        }
        #pragma unroll
        for (int o = 16; o; o >>= 1) sum += __shfl_xor(sum, o, 32);
        const float inv = 1.0f / sum;
        for (int i = lane; i < TCTX; i += 32) sc_lds[i*4 + qr] *= inv;
    }
    __syncthreads();

    // ---------------- Phase 3: out^T = V^T (128x2048) . P^T (2048x4) --------
    const int d0 = wave * 16;
    const int dl = d0 + l15;                      // A-fragment row (M)
    v8f c = {};
    cur = 0;
    for (int t0 = 0; t0 < TCTX; t0 += 64) {
        __syncthreads();
        if (wave == 0) {
            if (t0 + 64 < TCTX) {
                tdm_load_2d(vc0 + (size_t)(t0 + 64) * 2 * DIM,
                            cur ? vb0 : vb1, 128, 64, 2 * DIM, 6);
                __builtin_amdgcn_s_wait_tensorcnt(1);
            } else {
                __builtin_amdgcn_s_wait_tensorcnt(0);
            }
            float* vt = cur ? vbuf1 : vbuf0;
            const int lo = (t0 > sp) ? t0 : sp;
            const int hi = ((t0 + 64) < (sp + SEQ)) ? (t0 + 64) : (sp + SEQ);
            for (int t = lo; t < hi; ++t) {
                float4 vv = *(const float4*)(vf0 + (size_t)(t - sp) * KVDIM + lane*4);
                *(float4*)(vt + (t - t0) * KPITCH + lane*4) = vv;
            }
        }
        __syncthreads();
        const float* vt = cur ? vbuf1 : vbuf0;
        #pragma unroll
        for (int t = 0; t < 64; t += 4) {
            const int ta = t + dsel;              // K index: 2*(lane/16)+{0,1}
            v2f a  = { vt[ta * KPITCH + dl], vt[(ta + 1) * KPITCH + dl] };
            v2f bv = { sc_lds[(t0 + ta) * 4 + l15], sc_lds[(t0 + ta + 1) * 4 + l15] };
            c = wmma_f32(a, bv, c);
        }
        cur ^= 1;
    }
    if (l15 < SEQ) {                              // valid output columns
        const int q  = l15;
        const int db = d0 + 8 * (lane >> 4);
        #pragma unroll
        for (int r = 0; r < 8; ++r)
            attn_out[(size_t)(b*SEQ + q) * DIM + h * HDIM + db + r] = c[r];
    }
}

// ---------------------------------------------------------------------------
// Output projection: out(32x4096) = attn(32x4096) @ wo^T.  grid = 64.
// ---------------------------------------------------------------------------
__global__ __launch_bounds__(256) void out_gemm(
    const float* __restrict__ a_in, const float* __restrict__ wo,
    float* __restrict__ out)
{
    __shared__ float a_lds[MTOK * LPITCH];
    __shared__ float b_lds[2][64 * LPITCH];

    const int tid  = threadIdx.x;
    const int lane = tid & 31;
    const int wave = tid >> 5;
    const int mt   = wave >> 2, nt = wave & 3;
    const int n_base = blockIdx.x * 64;

    const float* wbase = wo + (size_t)n_base * DIM;
    const unsigned int ldsb0 = (unsigned int)(uintptr_t)&b_lds[0][0];
    const unsigned int ldsb1 = (unsigned int)(uintptr_t)&b_lds[1][0];

    const int arow = tid >> 3;
    const int acol = (tid & 7) * 8;
    const int afrag = (mt*16 + (lane & 15)) * LPITCH + 2*(lane >> 4);
    const int bfrag = (nt*16 + (lane & 15)) * LPITCH + 2*(lane >> 4);

    if (wave == 0)
        tdm_load_2d(wbase, ldsb0, 64, 64, DIM, 5);

    v8f acc = {};
    int cur = 0;
    for (int k0 = 0; k0 < DIM; k0 += 64) {
        __syncthreads();
        {
            const float4* s = (const float4*)(a_in + (size_t)arow * DIM + k0 + acol);
            float4 p0 = s[0], p1 = s[1];
            *(float4*)(a_lds + arow*LPITCH + acol)     = p0;
            *(float4*)(a_lds + arow*LPITCH + acol + 4) = p1;
            if (k0 + 64 < DIM)
                __builtin_prefetch(a_in + (size_t)arow * DIM + k0 + 64 + acol, 0, 1);
        }
        if (wave == 0) {
            if (k0 + 64 < DIM) {
                tdm_load_2d(wbase + k0 + 64, cur ? ldsb0 : ldsb1, 64, 64, DIM, 5);
                __builtin_amdgcn_s_wait_tensorcnt(1);
            } else {
                __builtin_amdgcn_s_wait_tensorcnt(0);
            }
        }
        __syncthreads();
        const float* bb = b_lds[cur];
        #pragma unroll
        for (int kk = 0; kk < 64; kk += 4) {
            v2f a = *(const v2f*)(a_lds + afrag + kk);
            v2f b = *(const v2f*)(bb    + bfrag + kk);
            acc = wmma_f32(a, b, acc);
        }
        cur ^= 1;
    }
    const int n  = n_base + nt*16 + (lane & 15);
    const int mb = mt*16 + 8*(lane >> 4);
    #pragma unroll
    for (int r = 0; r < 8; ++r)
        out[(size_t)(mb + r) * DIM + n] = acc[r];
}

// ---------------------------------------------------------------------------
extern "C" void kernel_launch(void* const* d_in, const int* in_sizes, int n_in,
                              void* d_out, int out_size, void* d_ws, size_t ws_size,
                              hipStream_t stream)
{
    const float* x         = (const float*)d_in[0];
    const int*   start_pos = (const int*)  d_in[1];
    const float* angles    = (const float*)d_in[2];
    const float* cache_k   = (const float*)d_in[3];
    const float* cache_v   = (const float*)d_in[4];
    const float* mask      = (const float*)d_in[5];
    const float* wq        = (const float*)d_in[6];
    const float* wk        = (const float*)d_in[7];
    const float* wv        = (const float*)d_in[8];
    const float* wo        = (const float*)d_in[9];
    const int*   layer_idx = (const int*)  d_in[10];

    float* ws      = (float*)d_ws;
    float* q_ws    = ws;                          // 32*4096
    float* k_ws    = q_ws + MTOK * DIM;           // 32*1024
    float* v_ws    = k_ws + MTOK * KVDIM;         // 32*1024
    float* attn_ws = v_ws + MTOK * KVDIM;         // 32*4096

    qkv_gemm <<<NTOT / 64, 256, 0, stream>>>(x, wq, wk, wv, q_ws, k_ws, v_ws);
    rope_kernel<<<(MTOK * (NHEAD + NKVH) * (HDIM/2) + 255) / 256, 256, 0, stream>>>(
        q_ws, k_ws, angles);
    attn_kernel<<<BATCH * NHEAD, 256, 0, stream>>>(q_ws, k_ws, v_ws, cache_k,
        cache_v, mask, start_pos, layer_idx, attn_ws);
    out_gemm <<<DIM / 64, 256, 0, stream>>>(attn_ws, wo, (float*)d_out);
}